// PageEncoder_5944234738188
// MI455X (gfx1250) — compile-verified
//
#include <hip/hip_runtime.h>

// Problem constants (match reference)
#define B_  8
#define L_  8192
#define D_  1024
#define N_  256
#define P_  32
#define CHUNK   128        // d-columns staged per LDS tile
#define STRIDE  144        // padded LDS row length in floats (2*144 % 64 == 32 -> disjoint bank windows)

static constexpr float W_MEAN = 0.7f;
static constexpr float W_MAX  = 0.3f;
static constexpr float NEG_FILL = -10000.0f;

typedef __attribute__((ext_vector_type(2))) float v2f;
typedef __attribute__((ext_vector_type(8))) float v8f;

// ---- CDNA5 feature detection (compile-safe on both toolchains) ----
#define USE_ASYNC 0
#define USE_WMMA  0
#if defined(__has_builtin)
#if __has_builtin(__builtin_amdgcn_global_load_async_to_lds_b128) && \
    __has_builtin(__builtin_amdgcn_s_wait_asynccnt)
#undef USE_ASYNC
#define USE_ASYNC 1
#endif
#if __has_builtin(__builtin_amdgcn_wmma_f32_16x16x4_f32)
#undef USE_WMMA
#define USE_WMMA 1
#endif
#endif

// Async-LDS builtin signature (probe-confirmed round 1): (int4 AS1*, int4 AS3*, imm, imm)
typedef int v4i_ __attribute__((vector_size(16)));
typedef __attribute__((address_space(1))) v4i_* g4p_t;   // global int4*
typedef __attribute__((address_space(3))) v4i_* l4p_t;   // LDS int4*

__global__ __launch_bounds__(256) void page_encoder_kernel(
    const float* __restrict__ hs,        // [B, L, D]
    const int*   __restrict__ page_idx,  // [B, N, P]  (-1 = invalid)
    const int*   __restrict__ input_ids, // [B, L]
    const float* __restrict__ tlw,       // [B, L]
    const float* __restrict__ idf,       // [V]
    float*       __restrict__ out)       // [B, N, D]
{
    __shared__ float    tile[2][P_ * STRIDE];   // double-buffered 32 x 128 page tile
    __shared__ float    s_m[P_];                // validity mask per slot
    __shared__ float    s_w[P_];                // effective weight per slot
    __shared__ int      s_row[P_];              // clamped token row per slot
    __shared__ unsigned s_mbits;                // validity bitmask (bit p = slot p valid)
    __shared__ float    s_invU, s_invW;
    __shared__ int      s_low;

    const int bid = blockIdx.x;        // b*N + n
    const int b   = bid >> 8;          // N_ == 256
    const int t   = threadIdx.x;

    // ---- Phase 0: per-slot metadata (32 slots, exactly wave 0 -> full EXEC for ballot) ----
    if (t < P_) {
        const int idx   = page_idx[bid * P_ + t];
        const int valid = idx >= 0;
        const int row   = valid ? idx : 0;
        const int tok   = input_ids[b * L_ + row];
        s_m[t]   = valid ? 1.0f : 0.0f;
        s_w[t]   = valid ? idf[tok] * tlw[b * L_ + row] : 0.0f;
        s_row[t] = row;
        const unsigned bal = (unsigned)__ballot(valid);
        if (t == 0) s_mbits = bal;
    }
    __syncthreads();
    if (t == 0) {
        float ws = 0.0f, cnt = 0.0f;
        for (int p = 0; p < P_; ++p) { ws += s_w[p]; cnt += s_m[p]; }
        cnt = fmaxf(cnt, 1.0f);
        const int low = ws < 1e-4f;
        s_low  = low;
        s_invU = 1.0f / cnt;
        s_invW = low ? 0.0f : (1.0f / ws);
    }
    __syncthreads();

    const int      lane  = t & 31;
    const int      wv    = t >> 5;           // wave id 0..7
    const int      m16   = lane & 15;
    const int      hi    = lane >> 4;
    const unsigned mbits = s_mbits;

#if USE_WMMA
    // A-matrix operand per lane: A[m][k], m = lane&15, K = (vgpr) + 2*hi + 4*kc
    // row 0 = mask, row 1 = effective weights, rows 2..15 = 0
    float a0[8], a1[8];
#pragma unroll
    for (int kc = 0; kc < 8; ++kc) {
        const int k = 4 * kc + 2 * hi;
        float v0 = 0.0f, v1 = 0.0f;
        if (m16 == 0)      { v0 = s_m[k]; v1 = s_m[k + 1]; }
        else if (m16 == 1) { v0 = s_w[k]; v1 = s_w[k + 1]; }
        a0[kc] = v0; a1[kc] = v1;
    }
#endif

    // ---- Tile load assignment: 256 threads, thread -> (row, 16-float segment) ----
    const int    lr   = t >> 3;              // row 0..31
    const int    lc   = (t & 7) * 16;        // col segment 0,16,...,112
    const size_t gRow = ((size_t)b * L_ + (size_t)s_row[lr]) * D_;

    auto issue = [&](int c, int bi) {
        const float* g = hs + gRow + (size_t)c * CHUNK + lc;
        float*       l = &tile[bi][lr * STRIDE + lc];
#if USE_ASYNC
#pragma unroll
        for (int j = 0; j < 4; ++j)
            __builtin_amdgcn_global_load_async_to_lds_b128(
                (g4p_t)(g + 4 * j), (l4p_t)(l + 4 * j), 0, 0);
#else
#pragma unroll
        for (int j = 0; j < 4; ++j)
            *(float4*)(l + 4 * j) = *(const float4*)(g + 4 * j);
#endif
    };

    issue(0, 0);

    for (int c = 0; c < 8; ++c) {
        const int bi = c & 1;
        if (c < 7) issue(c + 1, bi ^ 1);     // prefetch next chunk (other buffer)
#if USE_ASYNC
        if (c < 7) __builtin_amdgcn_s_wait_asynccnt(4);  // drain current chunk, keep prefetch in flight
        else       __builtin_amdgcn_s_wait_asynccnt(0);
#endif
        __syncthreads();                      // all waves' chunk-c data visible

        const float* tb  = &tile[bi][0];
        const int    col = wv * 16 + m16;     // this wave's 16 columns within the 128-col chunk

        float mx = NEG_FILL;                  // masked max folded into the same LDS reads

        // ---- sum reductions (+ max from the same B-operand values) ----
#if USE_WMMA
        v8f acc = {};
#pragma unroll
        for (int kc = 0; kc < 8; ++kc) {
            const int r0 = 4 * kc + 2 * hi;   // B[k][n]: K = vgpr + 2*hi + 4*kc
            v2f a; a.x = a0[kc]; a.y = a1[kc];
            v2f bb; bb.x = tb[r0 * STRIDE + col];
                    bb.y = tb[(r0 + 1) * STRIDE + col];
            acc = __builtin_amdgcn_wmma_f32_16x16x4_f32(
                false, a, false, bb, (short)0, acc, false, false);
            const float c0 = ((mbits >> r0) & 1u)       ? bb.x : NEG_FILL;
            const float c1 = ((mbits >> (r0 + 1)) & 1u) ? bb.y : NEG_FILL;
            mx = fmaxf(mx, fmaxf(c0, c1));
        }
        const float usum = acc[0];   // M=0 row (lanes hi==0)
        const float wsum = acc[1];   // M=1 row (lanes hi==0)
#else
        float usum = 0.0f, wsum = 0.0f;
#pragma unroll
        for (int r = 0; r < P_; ++r) {
            const float v = tb[r * STRIDE + col];
            usum = fmaf(s_m[r], v, usum);
            wsum = fmaf(s_w[r], v, wsum);
            mx   = fmaxf(mx, ((mbits >> r) & 1u) ? v : NEG_FILL);
        }
#endif
        // lanes l and l+16 read the same column but complementary rows (≡{0,1} vs {2,3} mod 4):
        // combining the halves completes the 32-row max. (Idempotent in the fallback path.)
        mx = fmaxf(mx, __shfl_xor(mx, 16, 32));

        // ---- finalize & store (lanes 0..15 hold valid sums) ----
        const float mean = s_low ? usum * s_invU : wsum * s_invW;
        const float res  = W_MEAN * mean + W_MAX * mx;
        if (hi == 0)
            out[(size_t)bid * D_ + (size_t)c * CHUNK + col] = res;

        __syncthreads();                      // buffer bi free for chunk c+2 prefetch
    }
}

extern "C" void kernel_launch(void* const* d_in, const int* in_sizes, int n_in,
                              void* d_out, int out_size, void* d_ws, size_t ws_size,
                              hipStream_t stream) {
    (void)in_sizes; (void)n_in; (void)out_size; (void)d_ws; (void)ws_size;
    const float* hs   = (const float*)d_in[0];   // hidden_states  [B,L,D] f32
    const int*   pidx = (const int*)  d_in[1];   // page_indices   [B,N,P] int
    // d_in[2] = page_valid (bool) -- redundant: identical to (page_indices >= 0)
    const int*   ids  = (const int*)  d_in[3];   // input_ids      [B,L]   int
    const float* tw   = (const float*)d_in[4];   // token_level_weights [B,L] f32
    const float* idfw = (const float*)d_in[5];   // idf_weights    [V] f32
    float* out = (float*)d_out;                  // [B,N,D] f32

    dim3 grid(B_ * N_), block(256);
    hipLaunchKernelGGL(page_encoder_kernel, grid, block, 0, stream,
                       hs, pidx, ids, tw, idfw, out);
}